// SimpleRNN_37709812859255
// MI455X (gfx1250) — compile-verified
//
#include <hip/hip_runtime.h>
#include <hip/hip_bf16.h>

typedef float v2f __attribute__((ext_vector_type(2)));
typedef float v4f __attribute__((ext_vector_type(4)));
typedef float v8f __attribute__((ext_vector_type(8)));

// Problem dimensions (from reference)
constexpr int B = 4, T = 128, I = 32, H = 64;
constexpr int BT = B * T; // 512

// Flat float offsets of each output in d_out (concatenated in return order)
constexpr size_t OFF_FC  = 0;                                   // final_carry (B,H)
constexpr size_t OFF_OUT = OFF_FC  + (size_t)B * H;             // out (B,T,H)
constexpr size_t OFF_JWI = OFF_OUT + (size_t)B * T * H;         // J_Wi (B,T,H,I,H)
constexpr size_t OFF_JWH = OFF_JWI + (size_t)B * T * H * I * H; // J_Wh (B,T,H,H,H)
constexpr size_t OFF_JB  = OFF_JWH + (size_t)B * T * H * H * H; // J_b  (B,T,H,H)
constexpr size_t OFF_JH  = OFF_JB  + (size_t)B * T * H * H;     // J_h  (B,T,H,B,H)
constexpr size_t OUT_TOTAL = OFF_JH + (size_t)B * T * H * B * H; // 211,845,376 floats

// ---------------------------------------------------------------------------
// 1) Streaming zero-fill of the entire output: 847 MB of zeros dominates the
//    runtime (~36 us at 23.3 TB/s). float4 non-temporal stores bypass L2
//    (output is 4.4x the 192 MB L2; no reuse).
// ---------------------------------------------------------------------------
__global__ void zero_fill_nt(float* __restrict__ out, size_t n4) {
  size_t i = (size_t)blockIdx.x * blockDim.x + threadIdx.x;
  size_t stride = (size_t)gridDim.x * blockDim.x;
  v4f z = {0.f, 0.f, 0.f, 0.f};
  v4f* p = (v4f*)out;
  for (; i < n4; i += stride) {
    __builtin_nontemporal_store(z, p + i);
  }
}

// ---------------------------------------------------------------------------
// 2) Pre-GEMM: XWi[row, j] = sum_k X[row,k]*Wi[k,j] + bias[j]
//    (512x32)@(32x64) in f32 via V_WMMA_F32_16X16X4_F32 (keeps reference
//    precision). One wave per 16x16 tile; K=32 in 8 steps of 4.
//    A-frag (16x4, f32): lane L -> M=L%16; VGPRs {0,1} = K{0,1} for lanes
//    0-15, K{2,3} for lanes 16-31. B-frag (4x16) mirrors along N.
//    C/D: VGPR r -> M = r (+8 for lanes 16-31), N = lane%16.
// ---------------------------------------------------------------------------
__global__ void gemm_xwi_wmma(const float* __restrict__ X,
                              const float* __restrict__ Wi,
                              const float* __restrict__ bias,
                              float* __restrict__ XWi) {
  const int tile = blockIdx.x;       // 0..127
  const int tm = tile >> 2;          // row tile 0..31
  const int tn = tile & 3;           // col tile 0..3
  const int r0 = tm * 16, c0 = tn * 16;
  const int lane = threadIdx.x;      // wave32
  const int half = lane >> 4;        // 0 or 1
  const int l16  = lane & 15;

  v8f acc = {};
#pragma unroll
  for (int kk = 0; kk < 8; ++kk) {
    const int kb = kk * 4 + half * 2;
    v2f a, bm;
    a.x  = X[(r0 + l16) * I + kb];
    a.y  = X[(r0 + l16) * I + kb + 1];
    bm.x = Wi[kb * H + c0 + l16];
    bm.y = Wi[(kb + 1) * H + c0 + l16];
    acc = __builtin_amdgcn_wmma_f32_16x16x4_f32(
        /*neg_a=*/false, a, /*neg_b=*/false, bm,
        /*c_mod=*/(short)0, acc, /*reuse_a=*/false, /*reuse_b=*/false);
  }

  const int col = c0 + l16;
  const float bv = bias[col];
#pragma unroll
  for (int r = 0; r < 8; ++r) {
    const int row = r0 + r + half * 8;
    XWi[row * H + col] = acc[r] + bv;
  }
}

// ---------------------------------------------------------------------------
// 3) Sequential scan: one block, 256 threads = one (b,j) state each.
//    Wh lives in LDS (16 KB), h lives in LDS (1 KB). Emits out, final_carry,
//    and per-step g / h_prev to workspace for the Jacobian scatter.
// ---------------------------------------------------------------------------
__global__ void rnn_scan(const float* __restrict__ carry,
                         const float* __restrict__ Wh,
                         const float* __restrict__ XWi,
                         float* __restrict__ out,
                         float* __restrict__ g_ws,
                         float* __restrict__ hprev_ws) {
  __shared__ float WhS[H * H];
  __shared__ float hS[B * H];
  const int tid = threadIdx.x; // 0..255
  for (int k = tid; k < H * H; k += 256) WhS[k] = Wh[k];
  hS[tid] = carry[tid];        // B*H == 256
  __syncthreads();

  const int b = tid >> 6;
  const int j = tid & 63;
  float a = 0.f;
  for (int t = 0; t < T; ++t) {
    const float hp = hS[b * H + j];
    float pre = XWi[(b * T + t) * H + j]; // includes bias
#pragma unroll 8
    for (int m = 0; m < H; ++m) pre += hS[b * H + m] * WhS[m * H + j];
    a = tanhf(pre);
    const float g = 1.f - a * a;
    const size_t bt = (size_t)(b * T + t) * H + j;
    out[OFF_OUT + bt] = a;
    g_ws[bt]     = g;
    hprev_ws[bt] = hp;
    __syncthreads();
    hS[b * H + j] = a;
    __syncthreads();
  }
  out[OFF_FC + b * H + j] = a; // final carry
}

// ---------------------------------------------------------------------------
// 4) Diagonal scatter kernels (~20 MB of true nonzeros total).
// ---------------------------------------------------------------------------
__global__ void fill_jwi(const float* __restrict__ g, const float* __restrict__ X,
                         float* __restrict__ out) {
  const size_t total = (size_t)BT * H * I;
  size_t idx = (size_t)blockIdx.x * blockDim.x + threadIdx.x;
  if (idx >= total) return;
  const int i = (int)(idx % I);
  const int j = (int)((idx / I) % H);
  const size_t bt = idx / ((size_t)I * H); // 0..511
  const float v = g[bt * H + j] * X[bt * I + i];
  out[OFF_JWI + ((bt * H + j) * I + i) * H + j] = v;
}

__global__ void fill_jwh(const float* __restrict__ g, const float* __restrict__ hprev,
                         float* __restrict__ out) {
  const size_t total = (size_t)BT * H * H;
  size_t idx = (size_t)blockIdx.x * blockDim.x + threadIdx.x;
  if (idx >= total) return;
  const int m = (int)(idx % H);
  const int j = (int)((idx / H) % H);
  const size_t bt = idx / ((size_t)H * H);
  const float v = g[bt * H + j] * hprev[bt * H + m];
  out[OFF_JWH + ((bt * H + j) * H + m) * H + j] = v;
}

__global__ void fill_jb(const float* __restrict__ g, float* __restrict__ out) {
  const size_t total = (size_t)BT * H;
  size_t idx = (size_t)blockIdx.x * blockDim.x + threadIdx.x;
  if (idx >= total) return;
  const int j = (int)(idx % H);
  const size_t bt = idx / H;
  out[OFF_JB + (bt * H + j) * H + j] = g[idx];
}

__global__ void fill_jh(const float* __restrict__ g, const float* __restrict__ Wh,
                        float* __restrict__ out) {
  const size_t total = (size_t)BT * H * H;
  size_t idx = (size_t)blockIdx.x * blockDim.x + threadIdx.x;
  if (idx >= total) return;
  const int m = (int)(idx % H);              // contiguous store dim
  const int j = (int)((idx / H) % H);
  const size_t bt = idx / ((size_t)H * H);
  const int b = (int)(bt / T);
  const float v = g[bt * H + j] * Wh[m * H + j];
  out[OFF_JH + ((bt * H + j) * (size_t)B + b) * H + m] = v;
}

// ---------------------------------------------------------------------------
extern "C" void kernel_launch(void* const* d_in, const int* in_sizes, int n_in,
                              void* d_out, int out_size, void* d_ws, size_t ws_size,
                              hipStream_t stream) {
  (void)in_sizes; (void)n_in; (void)out_size; (void)ws_size;
  const float* carry = (const float*)d_in[0]; // (B,H)
  const float* x     = (const float*)d_in[1]; // (B,T,I)
  const float* Wi    = (const float*)d_in[2]; // (I,H)
  const float* Wh    = (const float*)d_in[3]; // (H,H)
  const float* bias  = (const float*)d_in[4]; // (H,)
  float* out = (float*)d_out;

  // Workspace layout (floats): XWi[512*64] | g[B*T*H] | hprev[B*T*H]  (~384 KB)
  float* XWi      = (float*)d_ws;
  float* g_ws     = XWi + (size_t)BT * H;
  float* hprev_ws = g_ws + (size_t)BT * H;

  // 1) zero everything (bandwidth-bound phase)
  zero_fill_nt<<<4096, 256, 0, stream>>>(out, OUT_TOTAL / 4);

  // 2) X @ Wi + b via f32 WMMA: 128 tiles, one wave each
  gemm_xwi_wmma<<<(BT / 16) * (H / 16), 32, 0, stream>>>(x, Wi, bias, XWi);

  // 3) sequential recurrence
  rnn_scan<<<1, 256, 0, stream>>>(carry, Wh, XWi, out, g_ws, hprev_ws);

  // 4) scatter the structural-diagonal Jacobian entries
  {
    const size_t n = (size_t)BT * H * I;
    fill_jwi<<<(unsigned)((n + 255) / 256), 256, 0, stream>>>(g_ws, x, out);
  }
  {
    const size_t n = (size_t)BT * H * H;
    fill_jwh<<<(unsigned)((n + 255) / 256), 256, 0, stream>>>(g_ws, hprev_ws, out);
  }
  {
    const size_t n = (size_t)BT * H;
    fill_jb<<<(unsigned)((n + 255) / 256), 256, 0, stream>>>(g_ws, out);
  }
  {
    const size_t n = (size_t)BT * H * H;
    fill_jh<<<(unsigned)((n + 255) / 256), 256, 0, stream>>>(g_ws, Wh, out);
  }
}